// HyenaCascade_86646670229722
// MI455X (gfx1250) — compile-verified
//
#include <hip/hip_runtime.h>
#include <hip/hip_bf16.h>

// ---------------------------------------------------------------------------
// Hyena cascade for MI455X (gfx1250).
// FFT long-conv replaced by its exact SSM form (8 diagonal states/channel),
// evaluated as a chunked scan. Intra-chunk Toeplitz conv + carry injection
// run on V_WMMA_F32_16X16X4_F32 (6 WMMAs per wave). Output staged via LDS
// for 64B-contiguous stores.
// ---------------------------------------------------------------------------

#define B_   2
#define L_   8192
#define H_   2048          // HIDDEN
#define H3_  6144          // 3*HIDDEN
#define S_   8             // states per channel
#define CH_  16            // scan chunk length (== WMMA tile)
#define NC_  512           // L_/CH_ chunks per channel
#define NG_  32            // NC_/16 chunk-groups (one wave each)

typedef __attribute__((ext_vector_type(2))) float v2f;
typedef __attribute__((ext_vector_type(4))) float v4f;
typedef __attribute__((ext_vector_type(8))) float v8f;

// ---------------- Kernel 1: FIR(K=3) + bias + split + gate + chunk states --
// One thread = one (b, d, chunk of 16 t). Sliding-window registers give
// ~3.4 u-loads per output (vs 9 unfused); x1v/x2 stored as float4;
// chunk-local SSM end-states e computed inline (kills the old pass 2).
__global__ void __launch_bounds__(256)
k_front(const float* __restrict__ u, const float* __restrict__ w,
        const float* __restrict__ bias, const float* __restrict__ log_poles,
        float* __restrict__ x1v, float* __restrict__ x2s, float* __restrict__ e) {
  int idx = blockIdx.x * 256 + threadIdx.x;      // [0, B*H*NC) = 2^21
  int d = idx & (H_ - 1);                        // d fastest -> coalesced u
  int chunk = (idx >> 11) & (NC_ - 1);
  int b = idx >> 20;
  int head = d >> 7, r = d & 127;
  int c2 = head * 384 + r;                       // head-wise column split
  int c1 = c2 + 128;
  int cv = c2 + 256;
  int t0 = chunk * CH_;
  const float* ub = u + ((size_t)b * L_ + t0) * H3_;

  float w1a = w[c1*3], w1b = w[c1*3+1], w1c = w[c1*3+2], b1 = bias[c1];
  float wva = w[cv*3], wvb = w[cv*3+1], wvc = w[cv*3+2], bv = bias[cv];
  float w2a = w[c2*3], w2b = w[c2*3+1], w2c = w[c2*3+2], b2 = bias[c2];

  float u1m2 = 0.f, u1m1 = 0.f, uvm2 = 0.f, uvm1 = 0.f, u2m2 = 0.f, u2m1 = 0.f;
  if (chunk > 0) {                               // t0 >= 16 -> rows valid
    u1m1 = ub[c1 - H3_];  u1m2 = ub[c1 - 2 * H3_];
    uvm1 = ub[cv - H3_];  uvm2 = ub[cv - 2 * H3_];
    u2m1 = ub[c2 - H3_];  u2m2 = ub[c2 - 2 * H3_];
  }
  float p[S_], acc[S_];
#pragma unroll
  for (int s = 0; s < S_; ++s) { p[s] = __expf(log_poles[d * S_ + s]); acc[s] = 0.f; }

  float xl[CH_], x2l[CH_];
#pragma unroll
  for (int j = 0; j < CH_; ++j) {
    const float* up = ub + (size_t)j * H3_;
    float u1 = up[c1], uv = up[cv], u2 = up[c2];
    float z1 = fmaf(w1c, u1, fmaf(w1b, u1m1, fmaf(w1a, u1m2, b1)));
    float zv = fmaf(wvc, uv, fmaf(wvb, uvm1, fmaf(wva, uvm2, bv)));
    float z2 = fmaf(w2c, u2, fmaf(w2b, u2m1, fmaf(w2a, u2m2, b2)));
    float xv = z1 * zv;
    xl[j] = xv; x2l[j] = z2;
#pragma unroll
    for (int s = 0; s < S_; ++s) acc[s] = fmaf(p[s], acc[s], xv);
    u1m2 = u1m1; u1m1 = u1;  uvm2 = uvm1; uvm1 = uv;  u2m2 = u2m1; u2m1 = u2;
  }
  size_t o = ((size_t)b * H_ + d) * L_ + t0;     // 64B-aligned (t0 % 16 == 0)
#pragma unroll
  for (int j = 0; j < CH_; j += 4) {
    v4f v1 = {xl[j], xl[j+1], xl[j+2], xl[j+3]};
    v4f v2 = {x2l[j], x2l[j+1], x2l[j+2], x2l[j+3]};
    *(v4f*)(x1v + o + j) = v1;
    *(v4f*)(x2s + o + j) = v2;
  }
  size_t eb = (((size_t)b * H_ + d) * S_) * NC_ + chunk;
#pragma unroll
  for (int s = 0; s < S_; ++s) e[eb + (size_t)s * NC_] = acc[s];
}

// ---------------- Kernel 2: carry scan across chunks -----------------------
// carry[(b,d,s), i] = state entering chunk i
__global__ void __launch_bounds__(256)
k_carry_scan(const float* __restrict__ e, const float* __restrict__ log_poles,
             float* __restrict__ carry) {
  int idx = blockIdx.x * 256 + threadIdx.x;      // [0, B*H*S) = 2^15
  int s = idx & (S_ - 1);
  int d = (idx >> 3) & (H_ - 1);
  int b = idx >> 14;
  float p16 = __expf(log_poles[d * S_ + s] * (float)CH_);
  size_t base = (((size_t)b * H_ + d) * S_ + s) * NC_;
  float c = 0.f;
  for (int i = 0; i < NC_; ++i) {
    __builtin_prefetch(&e[base + i + 32], 0, 0); // global_prefetch_b8
    carry[base + i] = c;
    c = fmaf(p16, c, e[base + i]);
  }
}

// ---------------- Kernel 3: WMMA output pass -------------------------------
// One wave per (b, d, group of 16 chunks) == 256 outputs; 16-wave blocks own
// 16 consecutive d of the same 256-t window (LDS-staged coalesced stores).
//   Y[t,chunk] = sum_j T[t,j] X[j,chunk] + sum_s A2[t,s] C[s,chunk]
//   T[t,j] = h[t-j] (lower-tri Toeplitz), A2[t,s] = r_s p_s^{t+1}
// f32 WMMA layouts (ISA 7.12.2):
//   A 16x4: lane<16 -> {A[M=lane,K=0],A[M=lane,K=1]}, lane>=16 -> K=2,3
//   B 4x16 (mirrored): lane<16 -> {B[K=0,N=lane],B[K=1,N=lane]}, hi -> K=2,3
//   C/D:   lane<16: c[v]=D[M=v,N=lane]; lane>=16: c[v]=D[M=v+8,N=lane-16]
__global__ void __launch_bounds__(512)
k_wmma_out(const float* __restrict__ x1v, const float* __restrict__ x2s,
           const float* __restrict__ carry,
           const float* __restrict__ log_poles, const float* __restrict__ residues,
           const float* __restrict__ Dvec, float* __restrict__ out) {
  __shared__ float tile[16][260];                // [d_local][t_local], padded

  int wave = threadIdx.x >> 5;
  int lane = threadIdx.x & 31;
  int wid  = blockIdx.x * 16 + wave;             // [0, B*H*NG) = 2^17 waves
  int d = wid & (H_ - 1);                        // block: d0..d0+15, same g,b
  int g = (wid >> 11) & (NG_ - 1);
  int b = wid >> 16;
  int n  = lane & 15;                            // N column (chunk) & A row t
  int hi = lane >> 4;

  // Per-lane filter ingredients: q_s = r_s * p_s^n  (row t = n)
  float p[S_], q[S_], invp[S_], invp3[S_], wsc[S_];
#pragma unroll
  for (int s = 0; s < S_; ++s) {
    float lp = log_poles[d * S_ + s];
    p[s] = __expf(lp);
    float p2 = p[s] * p[s], p4 = p2 * p2, p8 = p4 * p4;
    float pt = 1.f;
    if (n & 1) pt *= p[s];
    if (n & 2) pt *= p2;
    if (n & 4) pt *= p4;
    if (n & 8) pt *= p8;
    q[s] = residues[d * S_ + s] * pt;
    invp[s] = 1.0f / p[s];
    float i2 = invp[s] * invp[s];
    invp3[s] = i2 * invp[s];
    wsc[s] = hi ? i2 : 1.0f;                     // p^{-c}, c starts at 2*hi
  }
  // Toeplitz entries this lane feeds: cols c = 4*kk + 2*hi + {0,1}, kk=0..3
  float Tv[8];
  int c = 2 * hi;
#pragma unroll
  for (int ci = 0; ci < 8; ++ci) {
    float a = 0.f;
#pragma unroll
    for (int s = 0; s < S_; ++s) a = fmaf(q[s], wsc[s], a);      // h[n-c]
    Tv[ci] = (n >= c) ? a : 0.f;
    if (ci & 1) { c += 3;
#pragma unroll
      for (int s = 0; s < S_; ++s) wsc[s] *= invp3[s];
    } else { c += 1;
#pragma unroll
      for (int s = 0; s < S_; ++s) wsc[s] *= invp[s];
    }
  }

  size_t chbase = ((size_t)b * H_ + d) * L_ + (size_t)g * 256;
  const float* xb = x1v + chbase + n * 16;                 // chunk n, 16 taps
  const float* cb = carry + (((size_t)b * H_ + d) * S_) * NC_ + g * 16 + n;

  v8f acc = {};
  // Toeplitz: K = 16 via 4 chained K=4 WMMAs
#pragma unroll
  for (int kk = 0; kk < 4; ++kk) {
    v2f a; a.x = Tv[2 * kk]; a.y = Tv[2 * kk + 1];
    v2f bb = *(const v2f*)(xb + 4 * kk + 2 * hi);          // 8B aligned
    acc = __builtin_amdgcn_wmma_f32_16x16x4_f32(false, a, false, bb,
                                                (short)0, acc, false, false);
  }
  // Carry injection: K = 8 via 2 chained K=4 WMMAs; A2[t,s] = q_s * p_s.
  // Branchless (cndmask selects + arithmetic addressing): no exec changes.
#pragma unroll
  for (int kk = 0; kk < 2; ++kk) {
    v2f a;
    a.x = hi ? q[4*kk+2] * p[4*kk+2] : q[4*kk+0] * p[4*kk+0];
    a.y = hi ? q[4*kk+3] * p[4*kk+3] : q[4*kk+1] * p[4*kk+1];
    const float* cbk = cb + (size_t)(4 * kk + 2 * hi) * NC_;
    v2f bb; bb.x = cbk[0]; bb.y = cbk[NC_];
    acc = __builtin_amdgcn_wmma_f32_16x16x4_f32(false, a, false, bb,
                                                (short)0, acc, false, false);
  }

  // Epilogue: y = (y + x1v*D) * x2, staged to LDS (ds_store_b128 x2 / lane)
  float Dd = Dvec[d];
  int tt = n * 16 + 8 * hi;                      // 8 consecutive local t
  size_t tloc = (size_t)g * 256 + tt;
  const float* x1p = x1v + ((size_t)b * H_ + d) * L_ + tloc;
  const float* x2p = x2s + ((size_t)b * H_ + d) * L_ + tloc;
  int dl = wid & 15;                             // == wave
  v4f y0, y1;
#pragma unroll
  for (int v = 0; v < 4; ++v) y0[v] = (acc[v]     + x1p[v]     * Dd) * x2p[v];
#pragma unroll
  for (int v = 0; v < 4; ++v) y1[v] = (acc[v + 4] + x1p[v + 4] * Dd) * x2p[v + 4];
  *(v4f*)&tile[dl][tt]     = y0;
  *(v4f*)&tile[dl][tt + 4] = y1;
  __syncthreads();

  // Cooperative transposed store: thread pair per t-row -> 64B contiguous
  int t = threadIdx.x >> 1, half = threadIdx.x & 1;
  int wid0 = blockIdx.x * 16;
  int d0 = wid0 & (H_ - 1);
  int g0 = (wid0 >> 11) & (NG_ - 1);
  int b0 = wid0 >> 16;
  float* op = out + ((size_t)b0 * L_ + (size_t)g0 * 256 + t) * H_ + d0 + half * 8;
  v4f o0 = {tile[half*8+0][t], tile[half*8+1][t], tile[half*8+2][t], tile[half*8+3][t]};
  v4f o1 = {tile[half*8+4][t], tile[half*8+5][t], tile[half*8+6][t], tile[half*8+7][t]};
  *(v4f*)(op)     = o0;
  *(v4f*)(op + 4) = o1;
}

// ---------------------------------------------------------------------------
extern "C" void kernel_launch(void* const* d_in, const int* in_sizes, int n_in,
                              void* d_out, int out_size, void* d_ws, size_t ws_size,
                              hipStream_t stream) {
  (void)in_sizes; (void)n_in; (void)out_size; (void)ws_size;
  const float* u         = (const float*)d_in[0];
  const float* sfw       = (const float*)d_in[1];
  const float* sfb       = (const float*)d_in[2];
  const float* log_poles = (const float*)d_in[3];
  const float* residues  = (const float*)d_in[4];
  const float* Dvec      = (const float*)d_in[5];
  float* out = (float*)d_out;

  float* ws    = (float*)d_ws;                   // ~403 MB of f32 scratch
  float* x1v   = ws;                                             // B*H*L
  float* x2s   = x1v + (size_t)B_ * H_ * L_;                     // B*H*L
  float* e     = x2s + (size_t)B_ * H_ * L_;                     // B*H*S*NC
  float* carry = e   + (size_t)B_ * H_ * S_ * NC_;               // B*H*S*NC

  k_front      <<<(B_ * H_ * NC_) / 256, 256, 0, stream>>>(u, sfw, sfb, log_poles,
                                                           x1v, x2s, e);
  k_carry_scan <<<(B_ * H_ * S_) / 256, 256, 0, stream>>>(e, log_poles, carry);
  k_wmma_out   <<<(B_ * H_ * NG_) / 16, 512, 0, stream>>>(x1v, x2s, carry,
                                                          log_poles, residues,
                                                          Dvec, out);
}